// GINE_40458591929262
// MI455X (gfx1250) — compile-verified
//
#include <hip/hip_runtime.h>

// ---------------------------------------------------------------------------
// GINE GNN for MI455X (gfx1250, wave32).
// Node-MLP GEMMs use V_WMMA_F32_16X16X4_F32 (full fp32 matrix pipe, no
// precision loss vs. the fp32 reference). Edge scatter + pooling are the
// memory-bound parts and use plain atomics (L2-resident working set).
// ---------------------------------------------------------------------------

typedef __attribute__((ext_vector_type(2))) float v2f;
typedef __attribute__((ext_vector_type(8))) float v8f;

#define FD 64            // feature dim D
#define NLAYER 5
#define BN_EPS 1e-5f

// ----------------------------- utility: zero -------------------------------
__global__ void zero_f32_kernel(float* __restrict__ p, long long n) {
  long long i = (long long)blockIdx.x * blockDim.x + threadIdx.x;
  if (i < n) p[i] = 0.0f;
}

// ------------------------- edge gather/scatter ------------------------------
// msg = relu(h[src] + (edge_attr * edgeW + edgeB)); aggr[dst] += msg
// One thread handles one edge x 4 features (float4 gather, 4 f32 atomics).
__global__ void edge_scatter_kernel(const float* __restrict__ h,
                                    const int* __restrict__ src,
                                    const int* __restrict__ dst,
                                    const float* __restrict__ eattr,
                                    const float* __restrict__ eW,   // [64]
                                    const float* __restrict__ eB,   // [64]
                                    float* __restrict__ aggr,
                                    long long E) {
  long long gid = (long long)blockIdx.x * blockDim.x + threadIdx.x;
  long long j = gid >> 4;               // edge id
  if (j >= E) return;
  int q = (int)(gid & 15) << 2;         // feature quad offset
  long long s = (long long)src[j];
  long long d = (long long)dst[j];
  float ea = eattr[j];
  const float4 hv = *(const float4*)(h + s * FD + q);
  const float4 ew = *(const float4*)(eW + q);
  const float4 eb = *(const float4*)(eB + q);
  float m0 = fmaxf(hv.x + ea * ew.x + eb.x, 0.0f);
  float m1 = fmaxf(hv.y + ea * ew.y + eb.y, 0.0f);
  float m2 = fmaxf(hv.z + ea * ew.z + eb.z, 0.0f);
  float m3 = fmaxf(hv.w + ea * ew.w + eb.w, 0.0f);
  float* a = aggr + d * FD + q;
  atomicAdd(a + 0, m0);
  atomicAdd(a + 1, m1);
  atomicAdd(a + 2, m2);
  atomicAdd(a + 3, m3);
}

// ----------------------- fused node update (WMMA) ---------------------------
// z  = (1+eps)*h + aggr                       [16 x 64] tile per block
// z1 = relu(BN(z @ W1 + b1))                  via v_wmma_f32_16x16x4_f32
// h' = relu(z1 @ W2 + b2)
// Block = 128 threads = 4 waves; wave w owns output columns [16w, 16w+16).
__global__ __launch_bounds__(128)
void node_update_kernel(const float* __restrict__ hin,
                        const float* __restrict__ aggr,
                        const float* __restrict__ epsp,   // &eps[layer]
                        const float* __restrict__ W1,     // [64][64] row-major
                        const float* __restrict__ b1,
                        const float* __restrict__ bng,
                        const float* __restrict__ bnb,
                        const float* __restrict__ W2,
                        const float* __restrict__ b2,
                        float* __restrict__ hout,
                        int N) {
  __shared__ float As[16][FD + 4];   // +4 pad: avoid 16-way LDS bank conflict
  __shared__ float Zs[16][FD + 4];

  const int t    = threadIdx.x;
  const int wv   = t >> 5;          // wave 0..3
  const int lane = t & 31;
  const int hl   = lane >> 4;       // half-wave: 0 -> K{0,1}, 1 -> K{2,3}
  const int lr   = lane & 15;
  const int n0   = wv * 16;         // this wave's output-column base
  const int tile = blockIdx.x;
  const float eps1    = 1.0f + epsp[0];
  const float inv_std = rsqrtf(1.0f + BN_EPS);   // BN eval: mean=0, var=1

  // Stage z = (1+eps)*h + aggr for 16 rows of this tile.
  for (int idx = t; idx < 16 * FD; idx += 128) {
    int m = idx >> 6, k = idx & 63;
    long long row = (long long)tile * 16 + m;
    float v = 0.0f;
    if (row < N) v = eps1 * hin[row * FD + k] + aggr[row * FD + k];
    As[m][k] = v;
  }
  __syncthreads();

  // ---- GEMM1: acc = z @ W1 (K=64 via 16x WMMA_F32_16X16X4_F32) ----
  // A frag (16x4 f32): lane<16 -> A[lr][kb+{0,1}], lane>=16 -> A[lr][kb+{2,3}]
  // B frag (4x16 f32): mirrored, N = n0+lr across lanes.
  v8f acc = {};
#pragma unroll
  for (int kb = 0; kb < FD; kb += 4) {
    int ka = kb + 2 * hl;
    v2f a, b;
    a.x = As[lr][ka];
    a.y = As[lr][ka + 1];
    b.x = W1[ka * FD + n0 + lr];
    b.y = W1[(ka + 1) * FD + n0 + lr];
    acc = __builtin_amdgcn_wmma_f32_16x16x4_f32(
        /*neg_a=*/false, a, /*neg_b=*/false, b,
        /*c_mod=*/(short)0, acc, /*reuse_a=*/false, /*reuse_b=*/false);
  }
  // Epilogue 1: + b1, BN scale/shift, relu -> Zs  (C/D layout: M=r+8*hl, N=n0+lr)
#pragma unroll
  for (int r = 0; r < 8; ++r) {
    int m = r + 8 * hl;
    int n = n0 + lr;
    float v = acc[r] + b1[n];
    v = v * (bng[n] * inv_std) + bnb[n];
    Zs[m][n] = fmaxf(v, 0.0f);
  }
  __syncthreads();

  // ---- GEMM2: acc2 = z1 @ W2 ----
  v8f acc2 = {};
#pragma unroll
  for (int kb = 0; kb < FD; kb += 4) {
    int ka = kb + 2 * hl;
    v2f a, b;
    a.x = Zs[lr][ka];
    a.y = Zs[lr][ka + 1];
    b.x = W2[ka * FD + n0 + lr];
    b.y = W2[(ka + 1) * FD + n0 + lr];
    acc2 = __builtin_amdgcn_wmma_f32_16x16x4_f32(
        false, a, false, b, (short)0, acc2, false, false);
  }
  // Epilogue 2: + b2, relu -> global
#pragma unroll
  for (int r = 0; r < 8; ++r) {
    int m = r + 8 * hl;
    long long row = (long long)tile * 16 + m;
    if (row < N) {
      int n = n0 + lr;
      hout[row * FD + n] = fmaxf(acc2[r] + b2[n], 0.0f);
    }
  }
}

// --------------------------- segment-max pooling ----------------------------
// pooled[g][l*64+d] = max over nodes n with batch[n]==g of h_l[n][d].
// Values are post-relu (>=0) so int-bits atomicMax with 0-init is exact.
__global__ void pool_max_kernel(const float* __restrict__ hbufs, // 5 stacked
                                const int* __restrict__ batch,
                                int* __restrict__ pooled,        // [G][320] bits
                                int N) {
  long long gid = (long long)blockIdx.x * blockDim.x + threadIdx.x;
  if (gid >= (long long)N * FD) return;
  int n = (int)(gid >> 6);
  int d = (int)(gid & 63);
  long long g = (long long)batch[n];
#pragma unroll
  for (int l = 0; l < NLAYER; ++l) {
    float v = hbufs[(long long)l * N * FD + (long long)n * FD + d];
    atomicMax(pooled + g * (NLAYER * FD) + l * FD + d, __float_as_int(v));
  }
}

// ------------------------------- readout MLP --------------------------------
__global__ void mlp1_kernel(const float* __restrict__ pooled, // [G][320]
                            const float* __restrict__ W,      // [320][256]
                            const float* __restrict__ b,      // [256]
                            float* __restrict__ out) {        // [G][256]
  int g = blockIdx.x, c = threadIdx.x;
  float acc = b[c];
  for (int k = 0; k < NLAYER * FD; ++k)
    acc += pooled[g * (NLAYER * FD) + k] * W[k * 256 + c];
  out[g * 256 + c] = fmaxf(acc, 0.0f);
}

__global__ void mlp2_kernel(const float* __restrict__ gmid,   // [G][256]
                            const float* __restrict__ W,      // [256][5]
                            const float* __restrict__ b,      // [5]
                            float* __restrict__ out, int G) { // [G][5]
  int t = blockIdx.x * blockDim.x + threadIdx.x;
  if (t >= G * 5) return;
  int g = t / 5, c = t % 5;
  float acc = b[c];
  for (int k = 0; k < 256; ++k)
    acc += gmid[g * 256 + k] * W[k * 5 + c];
  out[g * 5 + c] = acc;
}

// ------------------------------- launcher -----------------------------------
extern "C" void kernel_launch(void* const* d_in, const int* in_sizes, int n_in,
                              void* d_out, int out_size, void* d_ws, size_t ws_size,
                              hipStream_t stream) {
  const float* x     = (const float*)d_in[0];
  const int*   eidx  = (const int*)d_in[1];
  const float* eattr = (const float*)d_in[2];
  const int*   batch = (const int*)d_in[3];
  const float* eps   = (const float*)d_in[4];
  const float* edgeW = (const float*)d_in[5];
  const float* edgeB = (const float*)d_in[6];
  const float* W1    = (const float*)d_in[7];
  const float* b1    = (const float*)d_in[8];
  const float* bng   = (const float*)d_in[9];
  const float* bnb   = (const float*)d_in[10];
  const float* W2    = (const float*)d_in[11];
  const float* b2    = (const float*)d_in[12];
  const float* l1W   = (const float*)d_in[13];
  const float* l1b   = (const float*)d_in[14];
  const float* l2W   = (const float*)d_in[15];
  const float* l2b   = (const float*)d_in[16];

  const int       N = in_sizes[0] / FD;      // 100000
  const long long E = (long long)in_sizes[2]; // 3.2M (edge_attr is [E,1])
  const int       G = out_size / 5;          // 64

  // Workspace layout (floats):
  float* ws     = (float*)d_ws;
  float* hbuf   = ws;                                   // 5 * N*64 layer outs
  float* aggr   = hbuf + (long long)NLAYER * N * FD;    // N*64
  float* pooled = aggr + (long long)N * FD;             // G*320
  float* gmid   = pooled + (long long)G * NLAYER * FD;  // G*256

  const int* src = eidx;
  const int* dst = eidx + E;

  const long long nNode = (long long)N * FD;
  for (int l = 0; l < NLAYER; ++l) {
    const float* hin  = (l == 0) ? x : (hbuf + (long long)(l - 1) * N * FD);
    float*       hout = hbuf + (long long)l * N * FD;

    zero_f32_kernel<<<(unsigned)((nNode + 255) / 256), 256, 0, stream>>>(aggr, nNode);

    long long eThreads = E * 16;
    edge_scatter_kernel<<<(unsigned)((eThreads + 255) / 256), 256, 0, stream>>>(
        hin, src, dst, eattr, edgeW + l * FD, edgeB + l * FD, aggr, E);

    node_update_kernel<<<(unsigned)((N + 15) / 16), 128, 0, stream>>>(
        hin, aggr, eps + l,
        W1 + (long long)l * FD * FD, b1 + l * FD,
        bng + l * FD, bnb + l * FD,
        W2 + (long long)l * FD * FD, b2 + l * FD,
        hout, N);
  }

  const long long nPool = (long long)G * NLAYER * FD;
  zero_f32_kernel<<<(unsigned)((nPool + 255) / 256), 256, 0, stream>>>(pooled, nPool);
  pool_max_kernel<<<(unsigned)((nNode + 255) / 256), 256, 0, stream>>>(
      hbuf, batch, (int*)pooled, N);

  mlp1_kernel<<<G, 256, 0, stream>>>(pooled, l1W, l1b, gmid);
  mlp2_kernel<<<(G * 5 + 255) / 256, 256, 0, stream>>>(gmid, l2W, l2b, (float*)d_out, G);
}